// GCN_42322607735471
// MI455X (gfx1250) — compile-verified
//
#include <hip/hip_runtime.h>

typedef __attribute__((ext_vector_type(2))) float v2f;
typedef __attribute__((ext_vector_type(8))) float v8f;

#define HDIM 128      // IN == H == HH == 128
#define SCAN_BS 256

static inline int cdiv(long a, long b) { return (int)((a + b - 1) / b); }

// ---------------- utility ----------------
__global__ void k_fill_f(float* __restrict__ p, long n, float v) {
    long i = (long)blockIdx.x * blockDim.x + threadIdx.x;
    if (i < n) p[i] = v;
}
__global__ void k_fill_i(int* __restrict__ p, long n, int v) {
    long i = (long)blockIdx.x * blockDim.x + threadIdx.x;
    if (i < n) p[i] = v;
}
__global__ void k_copy_i(const int* __restrict__ a, int* __restrict__ b, int n) {
    int i = blockIdx.x * blockDim.x + threadIdx.x;
    if (i < n) b[i] = a[i];
}

// ---------------- degree -> dinv ----------------
__global__ void k_deg(const int* __restrict__ dst, float* __restrict__ deg, int E) {
    int e = blockIdx.x * blockDim.x + threadIdx.x;
    if (e < E) atomicAdd(&deg[dst[e]], 1.0f);
}
__global__ void k_rsqrt_inplace(float* __restrict__ d, int N) {
    int i = blockIdx.x * blockDim.x + threadIdx.x;
    if (i < N) d[i] = rsqrtf(d[i]);
}

// ---------------- CSR build (int histogram + 2-level scan + bucket fill) ----------------
__global__ void k_count(const int* __restrict__ dst, int* __restrict__ cnt, int E) {
    int e = blockIdx.x * blockDim.x + threadIdx.x;
    if (e < E) atomicAdd(&cnt[dst[e]], 1);
}

// per-block exclusive scan of cnt -> row_start ; block totals -> partial
__global__ void k_block_scan(const int* __restrict__ cnt, int* __restrict__ row_start,
                             int* __restrict__ partial, int N) {
    __shared__ int sm[SCAN_BS];
    int i = blockIdx.x * SCAN_BS + threadIdx.x;
    int v = (i < N) ? cnt[i] : 0;
    sm[threadIdx.x] = v;
    __syncthreads();
    for (int off = 1; off < SCAN_BS; off <<= 1) {
        int t = (threadIdx.x >= off) ? sm[threadIdx.x - off] : 0;
        __syncthreads();
        sm[threadIdx.x] += t;
        __syncthreads();
    }
    if (i < N) row_start[i] = sm[threadIdx.x] - v;      // exclusive
    if (threadIdx.x == SCAN_BS - 1) partial[blockIdx.x] = sm[SCAN_BS - 1];
}

// single block of 1024 threads: exclusive scan of block totals (nb <= 1024, N <= 262144)
__global__ void k_scan_partials(int* __restrict__ partial, int nb) {
    __shared__ int sm[1024];
    int v = ((int)threadIdx.x < nb) ? partial[threadIdx.x] : 0;
    sm[threadIdx.x] = v;
    __syncthreads();
    for (int off = 1; off < 1024; off <<= 1) {
        int t = (threadIdx.x >= (unsigned)off) ? sm[threadIdx.x - off] : 0;
        __syncthreads();
        sm[threadIdx.x] += t;
        __syncthreads();
    }
    if ((int)threadIdx.x < nb) partial[threadIdx.x] = sm[threadIdx.x] - v;  // exclusive
}

__global__ void k_add_offsets(int* __restrict__ row_start, const int* __restrict__ partial, int N) {
    int i = blockIdx.x * SCAN_BS + threadIdx.x;
    if (i < N) row_start[i] += partial[blockIdx.x];
}

__global__ void k_bucket(const int* __restrict__ src, const int* __restrict__ dst,
                         int* __restrict__ cursor, int* __restrict__ csr_src, int E) {
    int e = blockIdx.x * blockDim.x + threadIdx.x;
    if (e >= E) return;
    int d = dst[e];
    int pos = atomicAdd(&cursor[d], 1);
    csr_src[pos] = src[e];
}

// ---------------- dense GEMM via f32 WMMA: 4 M-tiles per wave, B loaded once ----------------
// C[N,128] = (RELU ? relu(A) : A) @ W[128,128]
// grid = ceil(N/64) blocks x 8 waves; wave (blk, warp) owns rows blk*64..+63, cols warp*16..+15
// RELU is a compile-time template parameter: no branch inside the WMMA loop.
template <bool RELU>
__global__ void k_gemm128_wmma(const float* __restrict__ A, const float* __restrict__ W,
                               float* __restrict__ C, int N) {
    const int lane = threadIdx.x & 31;
    const int warp = threadIdx.x >> 5;          // 0..7 -> column tile
    const int m    = lane & 15;
    const int hi   = lane >> 4;                 // 0: K={0,1}, 1: K={2,3}
    const int koff = hi * 2;
    const int col  = warp * 16 + m;
    const long base = (long)blockIdx.x * 64;

    const float* arow[4];
#pragma unroll
    for (int t = 0; t < 4; ++t) {
        long row = base + t * 16 + m;
        long rowc = row < N ? row : (long)N - 1;  // clamp keeps loads in-bounds, EXEC full
        arow[t] = A + rowc * HDIM;
    }

    v8f acc0 = {}, acc1 = {}, acc2 = {}, acc3 = {};
#pragma unroll 8
    for (int kb = 0; kb < HDIM; kb += 4) {
        v2f b; b.x = W[(kb + koff) * HDIM + col];
               b.y = W[(kb + koff + 1) * HDIM + col];
        v2f a0, a1, a2, a3;
        a0.x = arow[0][kb + koff]; a0.y = arow[0][kb + koff + 1];
        a1.x = arow[1][kb + koff]; a1.y = arow[1][kb + koff + 1];
        a2.x = arow[2][kb + koff]; a2.y = arow[2][kb + koff + 1];
        a3.x = arow[3][kb + koff]; a3.y = arow[3][kb + koff + 1];
        if (RELU) {
            a0.x = fmaxf(a0.x, 0.f); a0.y = fmaxf(a0.y, 0.f);
            a1.x = fmaxf(a1.x, 0.f); a1.y = fmaxf(a1.y, 0.f);
            a2.x = fmaxf(a2.x, 0.f); a2.y = fmaxf(a2.y, 0.f);
            a3.x = fmaxf(a3.x, 0.f); a3.y = fmaxf(a3.y, 0.f);
        }
        acc0 = __builtin_amdgcn_wmma_f32_16x16x4_f32(false, a0, false, b, (short)0, acc0, false, false);
        acc1 = __builtin_amdgcn_wmma_f32_16x16x4_f32(false, a1, false, b, (short)0, acc1, false, false);
        acc2 = __builtin_amdgcn_wmma_f32_16x16x4_f32(false, a2, false, b, (short)0, acc2, false, false);
        acc3 = __builtin_amdgcn_wmma_f32_16x16x4_f32(false, a3, false, b, (short)0, acc3, false, false);
    }

    // C/D layout: VGPR r -> M = r + 8*(lane>=16), N = lane&15
#pragma unroll
    for (int t = 0; t < 4; ++t) {
        const v8f acc = t == 0 ? acc0 : t == 1 ? acc1 : t == 2 ? acc2 : acc3;
        long row0 = base + t * 16 + hi * 8;
#pragma unroll
        for (int r = 0; r < 8; ++r) {
            long rr = row0 + r;
            if (rr < N) C[rr * HDIM + col] = acc[r];
        }
    }
}

// ---------------- pull-style aggregation (fused self-loop + bias), no FP atomics ----------------
// out[d,:] = bias + h[d,:]*dinv[d]^2 + sum_{s in N(d)} h[s,:]*dinv[s]*dinv[d]
// one wave per destination node; lane holds 4 channels (32*4 = 128)
__global__ void k_aggregate(const float* __restrict__ h, const int* __restrict__ csr_src,
                            const int* __restrict__ row_start, const int* __restrict__ cnt,
                            const float* __restrict__ dinv, const float* __restrict__ bias,
                            float* __restrict__ out, int N) {
    int d    = blockIdx.x * (blockDim.x >> 5) + (threadIdx.x >> 5);
    int lane = threadIdx.x & 31;
    if (d >= N) return;

    float dd = dinv[d];
    float4 bi = ((const float4*)bias)[lane];
    float4 hv = ((const float4*)(h + (long)d * HDIM))[lane];
    float ns = dd * dd;
    float4 acc;
    acc.x = bi.x + hv.x * ns;
    acc.y = bi.y + hv.y * ns;
    acc.z = bi.z + hv.z * ns;
    acc.w = bi.w + hv.w * ns;

    const int start = row_start[d];
    const int num   = cnt[d];
    for (int j = 0; j < num; ++j) {
        int s = csr_src[start + j];                  // uniform across wave
        float w = dinv[s] * dd;
        float4 u = ((const float4*)(h + (long)s * HDIM))[lane];
        acc.x += u.x * w;
        acc.y += u.y * w;
        acc.z += u.z * w;
        acc.w += u.w * w;
    }
    ((float4*)(out + (long)d * HDIM))[lane] = acc;
}

// ---------------- pooling ----------------
__global__ void k_cnt_pool(const int* __restrict__ batch, float* __restrict__ cnt, int N) {
    int i = blockIdx.x * blockDim.x + threadIdx.x;
    if (i < N) atomicAdd(&cnt[batch[i]], 1.0f);
}
__global__ void k_psum(const float* __restrict__ h, const int* __restrict__ batch,
                       float* __restrict__ sums, int N) {
    long idx = (long)blockIdx.x * blockDim.x + threadIdx.x;
    if (idx >= (long)N * HDIM) return;
    int i = (int)(idx >> 7);
    int c = (int)(idx & 127);
    atomicAdd(&sums[(long)batch[i] * HDIM + c], h[idx]);
}
__global__ void k_pdiv(const float* __restrict__ sums, const float* __restrict__ cnt,
                       float* __restrict__ g, int B) {
    long idx = (long)blockIdx.x * blockDim.x + threadIdx.x;
    if (idx >= (long)B * HDIM) return;
    int b = (int)(idx >> 7);
    float c = cnt[b];
    g[idx] = sums[idx] / (c > 1.0f ? c : 1.0f);
}

// ---------------- routed MLP head: one block (128 thr, 4 wave32) per sample ----------------
__global__ void k_head(const float* __restrict__ g, const int* __restrict__ r_target,
                       const float* __restrict__ Wh1, const float* __restrict__ bh1,
                       const float* __restrict__ Wh2, const float* __restrict__ bh2,
                       float* __restrict__ out, int B) {
    int b = blockIdx.x;
    int o = threadIdx.x;          // 0..127
    int t = r_target[b];
    const float* gv = g + (long)b * HDIM;
    const float* w1 = Wh1 + (long)t * HDIM * HDIM;

    float acc = bh1[(long)t * HDIM + o];
#pragma unroll 8
    for (int h = 0; h < HDIM; ++h)
        acc += gv[h] * w1[(long)h * HDIM + o];
    acc = acc > 0.f ? acc : 0.f;

    float p = acc * Wh2[(long)t * HDIM + o];
    for (int off = 16; off > 0; off >>= 1) p += __shfl_down(p, off, 32);
    __shared__ float red[4];
    if ((threadIdx.x & 31) == 0) red[threadIdx.x >> 5] = p;
    __syncthreads();
    if (threadIdx.x == 0)
        out[b] = red[0] + red[1] + red[2] + red[3] + bh2[t];
}

// ---------------- launcher ----------------
extern "C" void kernel_launch(void* const* d_in, const int* in_sizes, int n_in,
                              void* d_out, int out_size, void* d_ws, size_t ws_size,
                              hipStream_t stream) {
    const float* x    = (const float*)d_in[0];
    const int*   ei   = (const int*)d_in[1];
    const int*   bat  = (const int*)d_in[2];
    const int*   rt   = (const int*)d_in[3];
    const float* W1   = (const float*)d_in[4];
    const float* b1   = (const float*)d_in[5];
    const float* W2   = (const float*)d_in[6];
    const float* b2   = (const float*)d_in[7];
    const float* W3   = (const float*)d_in[8];
    const float* b3   = (const float*)d_in[9];
    const float* Wh1  = (const float*)d_in[10];
    const float* bh1  = (const float*)d_in[11];
    const float* Wh2  = (const float*)d_in[12];
    const float* bh2  = (const float*)d_in[13];

    const int N = in_sizes[0] / HDIM;
    const int E = in_sizes[1] / 2;
    const int B = in_sizes[3];
    const int* src = ei;
    const int* dst = ei + E;

    // workspace layout: float region then int region (all 4-byte; base is malloc-aligned)
    float* ws    = (float*)d_ws;
    float* buf0  = ws;                            // GEMM output h (per layer)  N*H
    float* buf1  = buf0 + (size_t)N * HDIM;       // layer in/out (ping)        N*H
    float* dinv  = buf1 + (size_t)N * HDIM;       // N
    float* sums  = dinv + N;                      // B*H
    float* cntf  = sums + (size_t)B * HDIM;       // B
    float* g     = cntf + B;                      // B*H
    int*   cnt_i = (int*)(g + (size_t)B * HDIM);  // N
    int*   rowst = cnt_i + N;                     // N
    int*   cursor= rowst + N;                     // N
    int*   part  = cursor + N;                    // 1024
    int*   csr   = part + 1024;                   // E

    const long NH = (long)N * HDIM;
    const int  nb = cdiv(N, SCAN_BS);             // 391 for N=100k (<=1024 required)

    // ---- degrees -> dinv ----
    k_fill_f<<<cdiv(N, 256), 256, 0, stream>>>(dinv, N, 1.0f);
    k_deg<<<cdiv(E, 256), 256, 0, stream>>>(dst, dinv, E);
    k_rsqrt_inplace<<<cdiv(N, 256), 256, 0, stream>>>(dinv, N);

    // ---- CSR build (once, reused by all 3 layers) ----
    k_fill_i<<<cdiv(N, 256), 256, 0, stream>>>(cnt_i, N, 0);
    k_count<<<cdiv(E, 256), 256, 0, stream>>>(dst, cnt_i, E);
    k_block_scan<<<nb, SCAN_BS, 0, stream>>>(cnt_i, rowst, part, N);
    k_scan_partials<<<1, 1024, 0, stream>>>(part, nb);
    k_add_offsets<<<nb, SCAN_BS, 0, stream>>>(rowst, part, N);
    k_copy_i<<<cdiv(N, 256), 256, 0, stream>>>(rowst, cursor, N);
    k_bucket<<<cdiv(E, 256), 256, 0, stream>>>(src, dst, cursor, csr, E);

    const int gemmBlocks = cdiv(N, 64);

    // ---- layer 1 ----
    k_gemm128_wmma<false><<<gemmBlocks, 256, 0, stream>>>(x, W1, buf0, N);
    k_aggregate<<<cdiv(N, 8), 256, 0, stream>>>(buf0, csr, rowst, cnt_i, dinv, b1, buf1, N);
    // ---- layer 2 (relu fused into GEMM input, compile-time) ----
    k_gemm128_wmma<true><<<gemmBlocks, 256, 0, stream>>>(buf1, W2, buf0, N);
    k_aggregate<<<cdiv(N, 8), 256, 0, stream>>>(buf0, csr, rowst, cnt_i, dinv, b2, buf1, N);
    // ---- layer 3 ----
    k_gemm128_wmma<true><<<gemmBlocks, 256, 0, stream>>>(buf1, W3, buf0, N);
    k_aggregate<<<cdiv(N, 8), 256, 0, stream>>>(buf0, csr, rowst, cnt_i, dinv, b3, buf1, N);

    // ---- global mean pool ----
    k_fill_f<<<cdiv((long)B * HDIM, 256), 256, 0, stream>>>(sums, (long)B * HDIM, 0.0f);
    k_fill_f<<<cdiv(B, 256), 256, 0, stream>>>(cntf, B, 0.0f);
    k_cnt_pool<<<cdiv(N, 256), 256, 0, stream>>>(bat, cntf, N);
    k_psum<<<cdiv(NH, 256), 256, 0, stream>>>(buf1, bat, sums, N);
    k_pdiv<<<cdiv((long)B * HDIM, 256), 256, 0, stream>>>(sums, cntf, g, B);

    // ---- routed task heads ----
    k_head<<<B, HDIM, 0, stream>>>(g, rt, Wh1, bh1, Wh2, bh2, (float*)d_out, B);
}